// FaceGCNLayer_4166118277559
// MI455X (gfx1250) — compile-verified
//
#include <hip/hip_runtime.h>
#include <hip/hip_bf16.h>

#define F_DIM 48

typedef float v2f __attribute__((ext_vector_type(2)));
typedef float v8f __attribute__((ext_vector_type(8)));

// ---------------------------------------------------------------------------
// Pass 1: zero the output accumulator (harness poisons d_out with 0xAA).
// ---------------------------------------------------------------------------
__global__ void zero_kernel(float* __restrict__ out, int n) {
    int i4 = (blockIdx.x * blockDim.x + threadIdx.x) * 4;
    if (i4 + 3 < n) {
        *reinterpret_cast<float4*>(out + i4) = make_float4(0.f, 0.f, 0.f, 0.f);
    } else {
        for (int k = i4; k < n; ++k) out[k] = 0.f;
    }
}

// ---------------------------------------------------------------------------
// Pass 2: edge scatter.  16 lanes per edge; lane f handles features
// f, f+16, f+32  ->  three fully-coalesced 64B segments covering the 192B
// feature row.  Non-returning global_atomic_add_f32 (fire-and-forget,
// STOREcnt only).  Lane 0 prefetches the gather row ~64 edges ahead
// (global_prefetch_b8) to hide L2 latency on the data-dependent gather.
// ---------------------------------------------------------------------------
__global__ void edge_scatter_kernel(const float* __restrict__ x,
                                    const int*   __restrict__ rows,
                                    const int*   __restrict__ cols,
                                    const float* __restrict__ vals,
                                    float* __restrict__ out,
                                    int nEdges) {
    int t = blockIdx.x * blockDim.x + threadIdx.x;
    int e = t >> 4;          // 16 lanes per edge
    int f = t & 15;
    if (e >= nEdges) return;

    // Software prefetch of a future edge's feature row into cache.
    int epf = e + 64;
    if (f == 0 && epf < nEdges) {
        int cpf = cols[epf];
        __builtin_prefetch(x + (long)cpf * F_DIM, 0, 1);
    }

    int   r = rows[e];
    int   c = cols[e];
    float v = vals[e];

    const float* xr   = x   + (long)c * F_DIM;
    float*       orow = out + (long)r * F_DIM;

    // Three coalesced loads + three non-returning f32 atomics per lane.
    atomicAdd(orow + f,      v * xr[f]);
    atomicAdd(orow + f + 16, v * xr[f + 16]);
    atomicAdd(orow + f + 32, v * xr[f + 32]);
}

// ---------------------------------------------------------------------------
// Pass 3: z = ax * diag(w)  as a real matrix product on the WMMA pipe.
// One wave (32 lanes) owns a 16-row x 48-col tile of the output, processed
// in-place as three 16x16 tiles.  Each tile is A(16x16) x diag(w)(16x16),
// computed as a chain of four V_WMMA_F32_16X16X4_F32 (K-chunks of 4),
// with the diagonal B chunk synthesized in registers.  Full f32 precision.
//
// ISA VGPR layouts (CDNA5 ISA):
//   A 16x4 f32 : lane = m + 16*h holds a.x=A(m,2h), a.y=A(m,2h+1)
//   B 4x16 f32 : lane = n + 16*h holds b.x=B(2h,n), b.y=B(2h+1,n)
//   C/D 16x16  : lane = n + 16*h, vgpr v holds D(v+8h, n)
// ---------------------------------------------------------------------------
__global__ void scale_wmma_kernel(float* __restrict__ out,
                                  const float* __restrict__ w,
                                  int nNodes) {
    int wave = (blockIdx.x * blockDim.x + threadIdx.x) >> 5;  // wave32
    int lane = threadIdx.x & 31;
    int rows0 = wave * 16;
    if (rows0 >= nNodes) return;   // uniform per wave: EXEC stays all-ones

    int n = lane & 15;   // output column within tile / A row m
    int h = lane >> 4;   // half-wave selector

#pragma unroll
    for (int j = 0; j < 3; ++j) {              // three 16-col blocks of F=48
        float wn = w[16 * j + n];              // diag value for column n
        v8f acc = {};
#pragma unroll
        for (int c4 = 0; c4 < 4; ++c4) {       // K chunks of 4
            int kbase = 16 * j + 4 * c4;

            // A chunk: rows rows0..rows0+15, cols kbase..kbase+3
            const float* ap = out + (long)(rows0 + n) * F_DIM + kbase + 2 * h;
            v2f a;
            a.x = ap[0];
            a.y = ap[1];

            // Diagonal B chunk: B(k, n) = (kbase + k == 16j + n) ? w : 0
            v2f b;
            b.x = (n == 4 * c4 + 2 * h)     ? wn : 0.f;
            b.y = (n == 4 * c4 + 2 * h + 1) ? wn : 0.f;

            acc = __builtin_amdgcn_wmma_f32_16x16x4_f32(
                /*neg_a=*/false, a, /*neg_b=*/false, b,
                /*c_mod=*/(short)0, acc, /*reuse_a=*/false, /*reuse_b=*/false);
        }
        // Scatter D back in-place (disjoint addresses vs. other tiles).
#pragma unroll
        for (int v = 0; v < 8; ++v) {
            out[(long)(rows0 + v + 8 * h) * F_DIM + 16 * j + n] = acc[v];
        }
    }
}

// ---------------------------------------------------------------------------
extern "C" void kernel_launch(void* const* d_in, const int* in_sizes, int n_in,
                              void* d_out, int out_size, void* d_ws, size_t ws_size,
                              hipStream_t stream) {
    const float* x    = (const float*)d_in[0];  // [1, N, 48] features
    const float* w    = (const float*)d_in[1];  // [1, 48]    weights
    const int*   rows = (const int*)  d_in[2];  // [E]
    const int*   cols = (const int*)  d_in[3];  // [E]
    const float* vals = (const float*)d_in[4];  // [E]
    float* out = (float*)d_out;                 // [N, 48]

    const int nEdges = in_sizes[2];
    const int nNodes = in_sizes[0] / F_DIM;

    // Pass 1: zero accumulator
    {
        int nquads = (out_size + 3) / 4;
        int blocks = (nquads + 255) / 256;
        zero_kernel<<<blocks, 256, 0, stream>>>(out, out_size);
    }

    // Pass 2: gather-scale-scatter over edges (atomic f32 adds)
    {
        long threads = (long)nEdges * 16;
        int blocks = (int)((threads + 255) / 256);
        edge_scatter_kernel<<<blocks, 256, 0, stream>>>(x, rows, cols, vals,
                                                        out, nEdges);
    }

    // Pass 3: in-place column scale as WMMA against diag(w)
    {
        int waves  = (nNodes + 15) / 16;        // one wave per 16-row tile
        int blocks = (waves + 7) / 8;           // 8 waves per 256-thread block
        scale_wmma_kernel<<<blocks, 256, 0, stream>>>(out, w, nNodes);
    }
}